// DigitCaps_24902220383029
// MI455X (gfx1250) — compile-verified
//
#include <hip/hip_runtime.h>
#include <hip/hip_bf16.h>

// Problem constants (from reference setup_inputs)
#define B_   256
#define R_   1152
#define C_   10
#define O_   16
#define I_   8
#define CO_  (C_ * O_)          // 160
#define MAX_ITERS 8

typedef __attribute__((ext_vector_type(2))) float v2f;
typedef __attribute__((ext_vector_type(8))) float v8f;

// ---------------------------------------------------------------------------
// Kernel 0: zero b_ij (routing logits) — must not rely on prior state.
// ---------------------------------------------------------------------------
__global__ void zero_f32(float* __restrict__ p, int n) {
    int i = blockIdx.x * blockDim.x + threadIdx.x;
    int stride = gridDim.x * blockDim.x;
    for (; i < n; i += stride) p[i] = 0.0f;
}

// ---------------------------------------------------------------------------
// Kernel 1: u_hat[b,r,c,o] = sum_i W[r,c,o,i] * x[b,r,i]
// Per r: GEMM  M=256 (batch) x N=160 (c*16+o) x K=8 (i), fp32 exact via
// V_WMMA_F32_16X16X4_F32 (two K=4 steps). One wave per 16x16 tile.
//
// A 16x4 fp32 layout : VGPR0 lanes0-15=K0 / lanes16-31=K2 ; VGPR1 K1/K3
// B 4x16 fp32 layout : VGPR0 lanes0-15=K0,N=lane / lanes16-31=K2 ; VGPR1 K1/K3
// D 16x16 fp32       : VGPRv lanes0-15 -> M=v, N=lane ; lanes16-31 -> M=v+8
// ---------------------------------------------------------------------------
__global__ void uhat_wmma_kernel(const float* __restrict__ x,
                                 const float* __restrict__ W,
                                 float* __restrict__ uhat) {
    const int wave = (blockIdx.x * blockDim.x + threadIdx.x) >> 5;
    const int lane = threadIdx.x & 31;

    int t = wave;
    const int nt = t % C_;  t /= C_;          // 10 N-tiles of 16 (c*16+o)
    const int mt = t % 16;  t /= 16;          // 16 M-tiles of 16 (batch)
    const int r  = t;
    if (r >= R_) return;                      // wave-uniform: EXEC stays all-1

    const int b0 = mt * 16;
    const int n0 = nt * 16;
    const int m     = b0 + (lane & 15);       // batch row this lane feeds A with
    const int kbase = (lane >> 4) * 2;        // 0 or 2

    // A fragments: x[m, r, i]
    const float* xp = x + (size_t)m * (R_ * I_) + (size_t)r * I_;
    const v2f a0 = *(const v2f*)(xp + kbase);       // K = {0,1} / {2,3}
    const v2f a1 = *(const v2f*)(xp + 4 + kbase);   // K = {4,5} / {6,7}

    // B fragments: W[r, co, i] with co = n0 + (lane&15)
    const float* wp = W + (size_t)r * (CO_ * I_) + (size_t)(n0 + (lane & 15)) * I_;
    const v2f bm0 = *(const v2f*)(wp + kbase);
    const v2f bm1 = *(const v2f*)(wp + 4 + kbase);

    v8f acc = {};
    acc = __builtin_amdgcn_wmma_f32_16x16x4_f32(false, a0, false, bm0,
                                                (short)0, acc, false, false);
    acc = __builtin_amdgcn_wmma_f32_16x16x4_f32(false, a1, false, bm1,
                                                (short)0, acc, false, false);

    // Store D: u_hat[row, r, col]
    const int row = b0 + (lane >> 4) * 8;
    const int col = n0 + (lane & 15);
    float* up = uhat + ((size_t)row * R_ + r) * CO_ + col;
    const size_t mstride = (size_t)R_ * CO_;  // +1 batch row
#pragma unroll
    for (int v = 0; v < 8; ++v) up[(size_t)v * mstride] = acc[v];
}

// ---------------------------------------------------------------------------
// Kernel 2 (per iteration): fused softmax(C) + weighted R-reduction + squash.
// One block per (b, c). Phase A stages c_ij[b, r, c] for all r in LDS.
// Phase B: 16 o-lanes x 16 r-groups accumulate s, tree-reduce, squash -> d_out.
// ---------------------------------------------------------------------------
__global__ void route_cs_kernel(const float* __restrict__ bij,
                                const float* __restrict__ uhat,
                                float* __restrict__ vout,
                                const int* __restrict__ nit, int it) {
    if (it >= nit[0]) return;

    const int b = blockIdx.x / C_;
    const int c = blockIdx.x % C_;
    const int tid = threadIdx.x;

    __shared__ float wsh[R_];     // c_ij[b, r, c] for this (b,c)
    __shared__ float accs[256];

    // Phase A: softmax over C per r (b_ij row is 10 floats)
    for (int r = tid; r < R_; r += 256) {
        const float* bp = bij + ((size_t)b * R_ + r) * C_;
        float bl[C_];
#pragma unroll
        for (int cc = 0; cc < C_; ++cc) bl[cc] = bp[cc];
        float mx = bl[0];
#pragma unroll
        for (int cc = 1; cc < C_; ++cc) mx = fmaxf(mx, bl[cc]);
        float den = 0.0f;
#pragma unroll
        for (int cc = 0; cc < C_; ++cc) den += __expf(bl[cc] - mx);
        wsh[r] = __expf(bl[c] - mx) / den;
    }
    __syncthreads();

    // Phase B: s[b,c,o] = sum_r w[r] * u_hat[b,r,c,o]  (coalesced 64B per r)
    const int o = tid & 15;
    const int g = tid >> 4;     // 16 r-groups
    float acc = 0.0f;
    const float* up = uhat + (size_t)b * R_ * CO_ + c * O_ + o;
    for (int r = g; r < R_; r += 16)
        acc = fmaf(wsh[r], up[(size_t)r * CO_], acc);
    accs[tid] = acc;
    __syncthreads();

    if (tid < 16) {
        float s = 0.0f;
#pragma unroll
        for (int gg = 0; gg < 16; ++gg) s += accs[gg * 16 + tid];
        // squash (elementwise, faithful to reference formula)
        const float sq = s * s;
        const float v = sq * s / ((1.0f + sq) * sqrtf(sq));
        vout[((size_t)b * C_ + c) * O_ + tid] = v;
    }
}

// ---------------------------------------------------------------------------
// Kernel 3 (per iteration, skipped on last): agreement update
// b_ij[b,r,c] += sum_o u_hat[b,r,c,o] * v[b,c,o].  One block per b, v in LDS.
// ---------------------------------------------------------------------------
__global__ void route_agree_kernel(float* __restrict__ bij,
                                   const float* __restrict__ uhat,
                                   const float* __restrict__ vout,
                                   const int* __restrict__ nit, int it) {
    if (it + 1 >= nit[0]) return;

    const int b = blockIdx.x;
    const int tid = threadIdx.x;

    __shared__ float4 vsh[CO_ / 4];                 // 40 float4 = v[b, :, :]
    if (tid < CO_ / 4)
        vsh[tid] = ((const float4*)vout)[(size_t)b * (CO_ / 4) + tid];
    __syncthreads();

    for (int idx = tid; idx < R_ * C_; idx += 256) {
        const int r = idx / C_;
        const int c = idx % C_;
        const float4* up =
            (const float4*)(uhat + ((size_t)b * R_ + r) * CO_ + c * O_);
        float d = 0.0f;
#pragma unroll
        for (int j = 0; j < 4; ++j) {
            const float4 u = up[j];
            const float4 v = vsh[c * 4 + j];
            d = fmaf(u.x, v.x, d);
            d = fmaf(u.y, v.y, d);
            d = fmaf(u.z, v.z, d);
            d = fmaf(u.w, v.w, d);
        }
        bij[(size_t)b * (R_ * C_) + idx] += d;
    }
}

// ---------------------------------------------------------------------------
extern "C" void kernel_launch(void* const* d_in, const int* in_sizes, int n_in,
                              void* d_out, int out_size, void* d_ws, size_t ws_size,
                              hipStream_t stream) {
    const float* x   = (const float*)d_in[0];   // [B, R, I]
    const float* W   = (const float*)d_in[1];   // [1, R, C, O, I]
    const int*   nit = (const int*)d_in[2];     // scalar (device)
    float* out = (float*)d_out;                 // [B, C, O, 1] == v flat

    // Workspace layout
    float* uhat = (float*)d_ws;                             // 188.7 MB
    float* bij  = uhat + (size_t)B_ * R_ * CO_;             //  11.8 MB

    // 1) zero routing logits
    zero_f32<<<1024, 256, 0, stream>>>(bij, B_ * R_ * C_);

    // 2) u_hat GEMM via fp32 WMMA: 1152*16*10 wave-tiles, 8 waves/block
    const int total_waves = R_ * 16 * C_;                   // 184320
    uhat_wmma_kernel<<<total_waves / 8, 256, 0, stream>>>(x, W, uhat);

    // 3) routing iterations (device-side guard against *nit)
    for (int it = 0; it < MAX_ITERS; ++it) {
        route_cs_kernel<<<B_ * C_, 256, 0, stream>>>(bij, uhat, out, nit, it);
        route_agree_kernel<<<B_, 256, 0, stream>>>(bij, uhat, out, nit, it);
    }
}